// QuantumPerceptron_70789650972699
// MI455X (gfx1250) — compile-verified
//
#include <hip/hip_runtime.h>

typedef float v2f __attribute__((ext_vector_type(2)));
typedef float v8f __attribute__((ext_vector_type(8)));

#define DIMN     512
#define HALF_DIM 256
#define BATCHN   4096
#define TSTEPS   9
#define OUTF     64

// CDNA5 async global->LDS staging (GLOBAL_LOAD_ASYNC_TO_LDS_*, ASYNCcnt).
// Set to 0 to fall back to synchronous load+ds_store staging.
#ifndef USE_ASYNC_LDS
#define USE_ASYNC_LDS 1
#endif

// LDS byte offset of a __shared__ object = low 32 bits of its generic address
// (ISA 10.2: LDS aperture -> LDS address is the low 32 bits; matches as3 lowering).
__device__ __forceinline__ unsigned lds_off(const void* p) {
  return (unsigned)(unsigned long long)p;
}

__device__ __forceinline__ void async_copy_b64(unsigned loff, const float* g) {
#if USE_ASYNC_LDS
  asm volatile("global_load_async_to_lds_b64 %0, %1, off"
               :
               : "v"(loff), "v"(g)
               : "memory");
#endif
}
__device__ __forceinline__ void async_copy_b32(unsigned loff, const float* g) {
#if USE_ASYNC_LDS
  asm volatile("global_load_async_to_lds_b32 %0, %1, off"
               :
               : "v"(loff), "v"(g)
               : "memory");
#endif
}
__device__ __forceinline__ void wait_async() {
#if USE_ASYNC_LDS
  asm volatile("s_wait_asynccnt 0x0" ::: "memory");
#endif
}

// ---------------------------------------------------------------------------
// WMMA f32 GEMM:  D = alpha * A * op(B) + beta * D + diag * I   (row-major)
// TM x TN tile per workgroup, TM/16 waves, each wave -> 16 rows x TN cols.
// K staged in DOUBLE-BUFFERED panels of 32 copied asynchronously to LDS while
// the current panel feeds V_WMMA_F32_16X16X4_F32; one barrier per panel.
//
// LDS layouts (per buffer):
//   As[TM][34]    : row stride 34 floats -> fragment reads bank-conflict free
//                   (34*r mod 64 distinct for r in 0..15), rows 8B-aligned.
//   Bs[16][TN+16] : K-PAIR layout, Bs[p][n] = {B[2p][n], B[2p+1][n]} (float2)
//                   so a B fragment is ONE aligned ds_load_b64 into an
//                   adjacent VGPR pair. v2f stride TN+16 => (2*TN+32) mod 64
//                   == 32, so half-waves p and p+1 use disjoint bank halves.
// ---------------------------------------------------------------------------
template <int TM, int TN, bool TRANSB>
__global__ __launch_bounds__(TM * 2) void wmma_gemm_kernel(
    const float* __restrict__ A, int lda,
    const float* __restrict__ B, int ldb,
    float* __restrict__ D, int ldc,
    int M, int N, int K, float alpha, float beta, float diag) {
  constexpr int NSUB = TN / 16;
  constexpr int BSTR = TN + 16;  // v2f units
  constexpr int NTHR = TM * 2;

  __shared__ float As[2][TM][34];
  __shared__ v2f   Bs[2][16][BSTR];

  const int tid  = threadIdx.x;
  const int lane = tid & 31;
  const int wv   = tid >> 5;   // wave 0..TM/16-1
  const int r    = lane & 15;  // row (A) / col (B) within 16
  const int half = lane >> 4;  // K-pair select {0,1} vs {2,3}

  const int bm = blockIdx.y * TM;
  const int bn = blockIdx.x * TN;

  auto stage = [&](int buf, int k0) {
    // ---- A panel: TM rows x 32 k, float2 granules ----
    for (int u = tid; u < TM * 16; u += NTHR) {
      int row = u >> 4, c = u & 15;
      const float* gp = &A[(size_t)(bm + row) * lda + k0 + 2 * c];
#if USE_ASYNC_LDS
      async_copy_b64(lds_off(&As[buf][row][2 * c]), gp);
#else
      *(v2f*)&As[buf][row][2 * c] = *(const v2f*)gp;
#endif
    }
    // ---- B panel: 16 k-pairs x TN cols ----
    if (TRANSB) {
      // logical B[k][n] = B[n*ldb + k]: k-pair contiguous in memory -> b64
      for (int u = tid; u < 16 * TN; u += NTHR) {
        int p = u & 15, n = u >> 4;
        const float* gp = &B[(size_t)(bn + n) * ldb + k0 + 2 * p];
#if USE_ASYNC_LDS
        async_copy_b64(lds_off(&Bs[buf][p][n]), gp);
#else
        Bs[buf][p][n] = *(const v2f*)gp;
#endif
      }
    } else {
      for (int u = tid; u < 16 * TN; u += NTHR) {
        int n = u % TN, p = u / TN;
        const float* g0 = &B[(size_t)(k0 + 2 * p) * ldb + bn + n];
#if USE_ASYNC_LDS
        unsigned l0 = lds_off(&Bs[buf][p][n]);
        async_copy_b32(l0, g0);
        async_copy_b32(l0 + 4u, g0 + ldb);
#else
        v2f t;
        t.x = g0[0];
        t.y = g0[ldb];
        Bs[buf][p][n] = t;
#endif
      }
    }
  };

  v8f acc[NSUB] = {};

  stage(0, 0);
  wait_async();
  __syncthreads();

  for (int k0 = 0; k0 < K; k0 += 32) {
    const int cur = (k0 >> 5) & 1;
    if (k0 + 32 < K) stage(cur ^ 1, k0 + 32);  // overlap copy with WMMA

#pragma unroll
    for (int kk = 0; kk < 32; kk += 4) {
      // A fragment 16x4: lanes 0-15 K={kk,kk+1}, lanes 16-31 K={kk+2,kk+3}
      v2f a = *(const v2f*)&As[cur][wv * 16 + r][kk + half * 2];
      const int p = (kk >> 1) + half;  // k-pair row for this half-wave
#pragma unroll
      for (int n = 0; n < NSUB; ++n) {
        v2f b = Bs[cur][p][n * 16 + r];  // one ds_load_b64
        acc[n] = __builtin_amdgcn_wmma_f32_16x16x4_f32(
            false, a, false, b, (short)0, acc[n], false, false);
      }
    }
    // next buffer's async copies complete + everyone done reading `cur`
    wait_async();
    __syncthreads();
  }

  // ---- epilogue: D layout: VGPR v -> M = v + 8*half, N = lane&15 ----
  const int row0 = bm + wv * 16 + half * 8;
#pragma unroll
  for (int n = 0; n < NSUB; ++n) {
    const int col = bn + n * 16 + r;
#pragma unroll
    for (int v = 0; v < 8; ++v) {
      const int row = row0 + v;
      const size_t idx = (size_t)row * ldc + col;
      float val = alpha * acc[n][v];
      if (beta != 0.0f) val += beta * D[idx];
      if (diag != 0.0f && row == col) val += diag;
      D[idx] = val;
    }
  }
}

// ---------------------------------------------------------------------------
// Elementwise helpers
// ---------------------------------------------------------------------------
__global__ void deinterleave_kernel(const float2* __restrict__ u,
                                    float* __restrict__ re,
                                    float* __restrict__ im, int n) {
  int i = blockIdx.x * blockDim.x + threadIdx.x;
  if (i < n) {
    float2 v = u[i];
    re[i] = v.x;
    im[i] = v.y;
  }
}

// out = s * 0.5 * (h + h^T)
__global__ void sym_scale_kernel(const float* __restrict__ h,
                                 float* __restrict__ out, float s) {
  int idx = blockIdx.x * blockDim.x + threadIdx.x;
  int i = idx >> 9, j = idx & 511;
  out[idx] = s * 0.5f * (h[i * DIMN + j] + h[j * DIMN + i]);
}

// Dst = a * X + c * I
__global__ void scale_diag_kernel(const float* __restrict__ X,
                                  float* __restrict__ Dst, float a, float c) {
  int idx = blockIdx.x * blockDim.x + threadIdx.x;
  int i = idx >> 9, j = idx & 511;
  Dst[idx] = a * X[idx] + ((i == j) ? c : 0.0f);
}

// Karatsuba prep: Asum = Ar + Ai ; Bsum = Br + Bi
__global__ void add_pairs_kernel(const float* __restrict__ Ar,
                                 const float* __restrict__ Ai,
                                 float* __restrict__ Asum,
                                 const float* __restrict__ Br,
                                 const float* __restrict__ Bi,
                                 float* __restrict__ Bsum) {
  int i = blockIdx.x * blockDim.x + threadIdx.x;
  Asum[i] = Ar[i] + Ai[i];
  Bsum[i] = Br[i] + Bi[i];
}

// Karatsuba finish: on entry Cr = T1 = Ar*Br, Ci = T3 = (Ar+Ai)(Br+Bi)
//   Cr <- T1 - T2 ; Ci <- T3 - T1 - T2     (T2 = Ai*Bi)
__global__ void karatsuba_combine_kernel(float* __restrict__ Cr,
                                         float* __restrict__ Ci,
                                         const float* __restrict__ T2) {
  int i = blockIdx.x * blockDim.x + threadIdx.x;
  float t1 = Cr[i], t2 = T2[i];
  Cr[i] = t1 - t2;
  Ci[i] = Ci[i] - t1 - t2;
}

// dst (512x256) = src[:, 0::2]   (kron(state,[1,0]) sees only even columns)
__global__ void pack_even_kernel(const float* __restrict__ src,
                                 float* __restrict__ dst) {
  int idx = blockIdx.x * blockDim.x + threadIdx.x;  // 512*256
  int i = idx >> 8, c = idx & 255;
  dst[idx] = src[(size_t)i * DIMN + 2 * c];
}

// r[b] = sum_i sign(i) * (Pr[i,b]^2 + Pi[i,b]^2), P is (512 x 4096)
__global__ void reduce_sign_kernel(const float* __restrict__ Pr,
                                   const float* __restrict__ Pi,
                                   float* __restrict__ rOut) {
  int b = blockIdx.x * blockDim.x + threadIdx.x;  // 0..4095
  float acc = 0.0f;
  for (int i = 0; i < DIMN; i += 2) {
    float xr = Pr[(size_t)i * BATCHN + b];
    float xi = Pi[(size_t)i * BATCHN + b];
    acc += xr * xr + xi * xi;
    float yr = Pr[(size_t)(i + 1) * BATCHN + b];
    float yi = Pi[(size_t)(i + 1) * BATCHN + b];
    acc -= yr * yr + yi * yi;
  }
  rOut[b] = acc;
}

// out[b,o] = relu(sum_t r[t,b] * W[o,t] + bias[o])
__global__ void head_kernel(const float* __restrict__ rbuf,
                            const float* __restrict__ W,
                            const float* __restrict__ bias,
                            float* __restrict__ out) {
  int idx = blockIdx.x * blockDim.x + threadIdx.x;  // BATCHN*OUTF
  int b = idx >> 6, o = idx & 63;
  float acc = bias[o];
#pragma unroll
  for (int t = 0; t < TSTEPS; ++t)
    acc += rbuf[t * BATCHN + b] * W[o * TSTEPS + t];
  out[idx] = acc > 0.0f ? acc : 0.0f;
}

// ---------------------------------------------------------------------------
// Orchestration
// ---------------------------------------------------------------------------
static inline void swapf(float*& a, float*& b) { float* t = a; a = b; b = t; }

extern "C" void kernel_launch(void* const* d_in, const int* in_sizes, int n_in,
                              void* d_out, int out_size, void* d_ws,
                              size_t ws_size, hipStream_t stream) {
  const float*  state = (const float*)d_in[0];   // (4096, 256) f32
  const float2* U     = (const float2*)d_in[1];  // (10, 512, 512) complex64
  const float*  ham   = (const float*)d_in[2];   // (512, 512) f32
  const float*  W     = (const float*)d_in[3];   // (64, 9) f32
  const float*  bias  = (const float*)d_in[4];   // (64,) f32
  float*        out   = (float*)d_out;           // (4096, 64) f32
  (void)in_sizes; (void)n_in; (void)out_size; (void)ws_size;

  const size_t MBf = (size_t)DIMN * DIMN;  // 262144 floats (1 MiB)
  float* w = (float*)d_ws;
  auto take = [&](size_t n) { float* p = w; w += n; return p; };

  float* Bth  = take(MBf);       // theta * H
  float* B2   = take(MBf);       // (theta*H)^2
  float* T0   = take(MBf);
  float* T1   = take(MBf);
  float* Ere  = take(MBf);       // Re exp
  float* Eim  = take(MBf);       // Im exp
  float* Qre  = take(MBf);       // squaring ping-pong
  float* Qim  = take(MBf);
  float* SumA = take(MBf);       // Karatsuba scratch
  float* SumB = take(MBf);
  float* Kt2  = take(MBf);
  float* Hs   = take(18 * MBf);  // E^(k+1), k=0..8, re/im
  float* UreB = take(10 * MBf);  // deinterleaved U
  float* UimB = take(10 * MBf);
  float* UfR  = take(MBf);
  float* UfI  = take(MBf);
  float* TmR  = take(MBf);
  float* TmI  = take(MBf);
  float* AeR  = take((size_t)DIMN * HALF_DIM);
  float* AeI  = take((size_t)DIMN * HALF_DIM);
  float* Pr   = take((size_t)DIMN * BATCHN);
  float* Pi   = take((size_t)DIMN * BATCHN);
  float* rb   = take((size_t)TSTEPS * BATCHN);

  const int EWB = (int)(MBf / 256);  // elementwise blocks over 512^2

  // 512x512x512 real GEMM, 64x64 tiles (grid 8x8 = 64 WGs for occupancy)
  auto gemm512 = [&](float* D, const float* A, const float* B, float alpha,
                     float beta, float diag) {
    hipLaunchKernelGGL((wmma_gemm_kernel<64, 64, false>),
                       dim3(DIMN / 64, DIMN / 64), dim3(128), 0, stream,
                       A, DIMN, B, DIMN, D, DIMN, DIMN, DIMN, DIMN,
                       alpha, beta, diag);
  };
  // 3-mult complex GEMM: C = A*B (all 512x512; dest must not alias sources)
  auto cgemm = [&](float* Cr, float* Ci, const float* Ar, const float* Ai,
                   const float* Br, const float* Bi) {
    add_pairs_kernel<<<EWB, 256, 0, stream>>>(Ar, Ai, SumA, Br, Bi, SumB);
    gemm512(Cr, Ar, Br, 1.0f, 0.0f, 0.0f);      // T1
    gemm512(Kt2, Ai, Bi, 1.0f, 0.0f, 0.0f);     // T2
    gemm512(Ci, SumA, SumB, 1.0f, 0.0f, 0.0f);  // T3
    karatsuba_combine_kernel<<<EWB, 256, 0, stream>>>(Cr, Ci, Kt2);
  };

  // 0. deinterleave complex U into planar re/im
  {
    int n = (int)(10 * MBf);
    deinterleave_kernel<<<(n + 255) / 256, 256, 0, stream>>>(U, UreB, UimB, n);
  }

  // 1. Bth = theta * 0.5*(ham + ham^T), theta = 0.1 / 2^4 (scaling step)
  const float theta = 0.1f / 16.0f;
  sym_scale_kernel<<<EWB, 256, 0, stream>>>(ham, Bth, theta);

  // 2. X = Bth^2
  gemm512(B2, Bth, Bth, 1.0f, 0.0f, 0.0f);

  // 3. cos(theta*H) via even Taylor (Horner in X):  -> Ere
  //    cos = I - X/2 + X^2/24 - X^3/720 + X^4/40320
  scale_diag_kernel<<<EWB, 256, 0, stream>>>(B2, T0, 1.0f / 40320.0f,
                                             -1.0f / 720.0f);
  gemm512(T1, B2, T0, 1.0f, 0.0f, 1.0f / 24.0f);
  gemm512(T0, B2, T1, 1.0f, 0.0f, -0.5f);
  gemm512(Ere, B2, T0, 1.0f, 0.0f, 1.0f);

  // 4. sin(theta*H) = Bth * Q, Q = I - X/6 + X^2/120 - X^3/5040 + X^4/362880
  //    Eim = -sin  (E0 = cos - i sin)
  scale_diag_kernel<<<EWB, 256, 0, stream>>>(B2, T0, 1.0f / 362880.0f,
                                             -1.0f / 5040.0f);
  gemm512(T1, B2, T0, 1.0f, 0.0f, 1.0f / 120.0f);
  gemm512(T0, B2, T1, 1.0f, 0.0f, -1.0f / 6.0f);
  gemm512(T1, B2, T0, 1.0f, 0.0f, 1.0f);
  gemm512(Eim, Bth, T1, -1.0f, 0.0f, 0.0f);

  // 5. square 4 times: E <- E^2  (ends with exp(-i*0.1*H))
  float *er = Ere, *ei = Eim, *qr = Qre, *qi = Qim;
  for (int s = 0; s < 4; ++s) {
    cgemm(qr, qi, er, ei, er, ei);
    swapf(er, qr);
    swapf(ei, qi);
  }

  // 6. powers: Hs[k] = E^(k+1)  (t_k = 0.1*(k+1))
  auto hsR = [&](int k) { return Hs + (size_t)k * 2 * MBf; };
  auto hsI = [&](int k) { return Hs + (size_t)k * 2 * MBf + MBf; };
  hipMemcpyAsync(hsR(0), er, MBf * sizeof(float), hipMemcpyDeviceToDevice,
                 stream);
  hipMemcpyAsync(hsI(0), ei, MBf * sizeof(float), hipMemcpyDeviceToDevice,
                 stream);
  for (int k = 1; k < TSTEPS; ++k)
    cgemm(hsR(k), hsI(k), hsR(k - 1), hsI(k - 1), er, ei);

  // 7. per-time circuit, apply to batch, signed |psi|^2 reduction
  for (int t = 0; t < TSTEPS; ++t) {
    float *ufR = UfR, *ufI = UfI, *tmR = TmR, *tmI = TmI;
    hipMemcpyAsync(ufR, UreB, MBf * sizeof(float), hipMemcpyDeviceToDevice,
                   stream);
    hipMemcpyAsync(ufI, UimB, MBf * sizeof(float), hipMemcpyDeviceToDevice,
                   stream);
    for (int l = 0; l < 5; ++l) {
      if (l > 0) {
        cgemm(tmR, tmI, UreB + (size_t)(2 * l) * MBf,
              UimB + (size_t)(2 * l) * MBf, ufR, ufI);
        swapf(ufR, tmR); swapf(ufI, tmI);
      }
      cgemm(tmR, tmI, hsR(t), hsI(t), ufR, ufI);
      swapf(ufR, tmR); swapf(ufI, tmI);
      cgemm(tmR, tmI, UreB + (size_t)(2 * l + 1) * MBf,
            UimB + (size_t)(2 * l + 1) * MBf, ufR, ufI);
      swapf(ufR, tmR); swapf(ufI, tmI);
    }
    // pack even columns, P = Ae * state^T  (512 x 4096, K = 256)
    pack_even_kernel<<<(DIMN * HALF_DIM) / 256, 256, 0, stream>>>(ufR, AeR);
    pack_even_kernel<<<(DIMN * HALF_DIM) / 256, 256, 0, stream>>>(ufI, AeI);
    hipLaunchKernelGGL((wmma_gemm_kernel<64, 128, true>),
                       dim3(BATCHN / 128, DIMN / 64), dim3(128), 0, stream,
                       AeR, HALF_DIM, state, HALF_DIM, Pr, BATCHN,
                       DIMN, BATCHN, HALF_DIM, 1.0f, 0.0f, 0.0f);
    hipLaunchKernelGGL((wmma_gemm_kernel<64, 128, true>),
                       dim3(BATCHN / 128, DIMN / 64), dim3(128), 0, stream,
                       AeI, HALF_DIM, state, HALF_DIM, Pi, BATCHN,
                       DIMN, BATCHN, HALF_DIM, 1.0f, 0.0f, 0.0f);
    reduce_sign_kernel<<<BATCHN / 256, 256, 0, stream>>>(
        Pr, Pi, rb + (size_t)t * BATCHN);
  }

  // 8. head: out = relu(r @ W^T + b)
  head_kernel<<<(BATCHN * OUTF) / 256, 256, 0, stream>>>(rb, W, bias, out);
}